// DistributionMaxPool_12352325943800
// MI455X (gfx1250) — compile-verified
//
#include <hip/hip_runtime.h>

typedef __attribute__((ext_vector_type(2))) float v2f;
typedef __attribute__((ext_vector_type(8))) float v8f;

#define EPSF 1e-8f
#define INV_SQRT2_F 0.70710678118654752440f
#define INV_SQRT_2PI_F 0.39894228040143267794f

__device__ __forceinline__ void max_gauss(float m1, float v1, float m2, float v2,
                                          float& mo, float& vo) {
    float alpha  = sqrtf(v1 + v2 + EPSF);
    float beta   = (m1 - m2) / alpha;
    float cdf_b  = 0.5f * (1.0f + erff(beta * INV_SQRT2_F));
    float cdf_nb = 0.5f * (1.0f + erff(-beta * INV_SQRT2_F));
    float pdf_b  = expf(-0.5f * beta * beta) * INV_SQRT_2PI_F;
    float mean   = m1 * cdf_b + m2 * cdf_nb + alpha * pdf_b;
    float var    = (v1 + m1 * m1) * cdf_b + (v2 + m2 * m2) * cdf_nb
                 + (m1 + m2) * alpha * pdf_b - mean * mean + EPSF;
    mo = mean; vo = var;
}

// Input  x  : (B=32, T=2, C=128, H=64, W=64) f32
// Output out: (B=32, T=2, C=128, HO=32, WO=32) f32
// Block: 128 threads = 4 waves. Block (i, b) handles one output row i of batch b.
// Wave w handles outputs j in [8w, 8w+8) via one 16-wide WMMA N-tile (w-range 16w..16w+15).
__global__ __launch_bounds__(128, 4)
void dist_maxpool_kernel(const float* __restrict__ x, float* __restrict__ out) {
    const int i    = blockIdx.x;        // output row, 0..31
    const int b    = blockIdx.y;        // batch, 0..31
    const int tid  = (int)threadIdx.x;
    const int wave = tid >> 5;
    const int lane = tid & 31;
    const int n    = lane & 15;         // N index inside WMMA tile
    const int hh   = lane >> 4;         // lane-half -> K-channel base offset 0 / 2

    const int j0 = wave * 8;
    const int w  = wave * 16 + n;       // input column = 2*j0 + n (contiguous!)

    const size_t CH = 4096;             // channel stride (floats) = 64*64
    // Per-(t,r) base pointers for this lane: x[b, t, c = 2*hh, 2i+r, w]
    const float* pm0 = x + (((size_t)(b * 2 + 0) * 128) + (size_t)(hh * 2)) * CH + (size_t)(2 * i + 0) * 64 + w;
    const float* pm1 = x + (((size_t)(b * 2 + 0) * 128) + (size_t)(hh * 2)) * CH + (size_t)(2 * i + 1) * 64 + w;
    const float* pv0 = x + (((size_t)(b * 2 + 1) * 128) + (size_t)(hh * 2)) * CH + (size_t)(2 * i + 0) * 64 + w;
    const float* pv1 = x + (((size_t)(b * 2 + 1) * 128) + (size_t)(hh * 2)) * CH + (size_t)(2 * i + 1) * 64 + w;

    const v2f ones = {1.0f, 1.0f};      // A = ones(16x4): layout-independent
    v8f am0 = {}; // channel-sum of means, row offset r=0
    v8f am1 = {}; // means, r=1
    v8f av0 = {}; // vars,  r=0
    v8f av1 = {}; // vars,  r=1

    // 128 channels, 4 per WMMA K-step
    for (int k = 0; k < 32; ++k) {
        const size_t off = (size_t)k * 4 * CH;
        v2f bm0, bm1, bv0, bv1;
        bm0.x = pm0[off]; bm0.y = pm0[off + CH];
        bm1.x = pm1[off]; bm1.y = pm1[off + CH];
        bv0.x = pv0[off]; bv0.y = pv0[off + CH];
        bv1.x = pv1[off]; bv1.y = pv1[off + CH];
        am0 = __builtin_amdgcn_wmma_f32_16x16x4_f32(false, ones, false, bm0, (short)0, am0, false, false);
        am1 = __builtin_amdgcn_wmma_f32_16x16x4_f32(false, ones, false, bm1, (short)0, am1, false, false);
        av0 = __builtin_amdgcn_wmma_f32_16x16x4_f32(false, ones, false, bv0, (short)0, av0, false, false);
        av1 = __builtin_amdgcn_wmma_f32_16x16x4_f32(false, ones, false, bv1, (short)0, av1, false, false);
    }

    // D is broadcast over M; element 0 of each accumulator in this lane is the
    // channel-sum for tap (r, s = n&1) at output column j = j0 + n/2.
    float sm0 = am0[0];
    float sm1 = am1[0];
    float sv0 = av0[0];
    float sv1 = av1[0];

    // Pair s=0 / s=1 taps across adjacent lanes.
    float om0 = __shfl_xor(sm0, 1, 32);
    float om1 = __shfl_xor(sm1, 1, 32);
    float ov0 = __shfl_xor(sv0, 1, 32);
    float ov1 = __shfl_xor(sv1, 1, 32);

    const bool even = ((n & 1) == 0);
    float m00 = even ? sm0 : om0;   // (r=0,s=0)
    float m01 = even ? om0 : sm0;   // (r=0,s=1)
    float m10 = even ? sm1 : om1;   // (r=1,s=0)
    float m11 = even ? om1 : sm1;   // (r=1,s=1)
    float v00 = even ? sv0 : ov0;
    float v01 = even ? ov0 : sv0;
    float v10 = even ? sv1 : ov1;
    float v11 = even ? ov1 : sv1;

    float hm1, hv1, hm2, hv2, fm, fv;
    max_gauss(m00, v00, m01, v01, hm1, hv1);
    max_gauss(m10, v10, m11, v11, hm2, hv2);
    max_gauss(hm1, hv1, hm2, hv2, fm, fv);

    __shared__ float smean[32];
    __shared__ float svar[32];
    if (lane < 16 && even) {
        int j = j0 + (n >> 1);
        smean[j] = fm;
        svar[j]  = fv;
    }
    __syncthreads();

    // Broadcast write: out[b, t, c, i, j] for t in {0,1}, c in 0..127, j in 0..31.
    // Each thread keeps j = lane fixed; tc = t*128 + c = it*4 + wave.
    const float mval = smean[lane];
    const float vval = svar[lane];
    const size_t obase = (size_t)b * 262144 + (size_t)i * 32 + (size_t)wave * 1024 + (size_t)lane;
    #pragma unroll 4
    for (int it = 0; it < 32; ++it) {            // tc = it*4 + wave < 128 -> means
        out[obase + (size_t)it * 4096] = mval;
    }
    #pragma unroll 4
    for (int it = 32; it < 64; ++it) {           // tc >= 128 -> vars
        out[obase + (size_t)it * 4096] = vval;
    }
}

extern "C" void kernel_launch(void* const* d_in, const int* in_sizes, int n_in,
                              void* d_out, int out_size, void* d_ws, size_t ws_size,
                              hipStream_t stream) {
    const float* x = (const float*)d_in[0];
    float* out = (float*)d_out;
    dim3 grid(32, 32);   // (i, b)
    dim3 block(128);
    hipLaunchKernelGGL(dist_maxpool_kernel, grid, block, 0, stream, x, out);
}